// resolution_attention_v3_20083267076713
// MI455X (gfx1250) — compile-verified
//
#include <hip/hip_runtime.h>
#include <hip/hip_bf16.h>
#include <math.h>

// ---------------------------------------------------------------------------
// MI455X (gfx1250) implementation.
//
// Math restructuring:
//   score_s = (Q @ Wk) . x_s   (+const that cancels in softmax)
//   ctx     = sum_s softmax_s * x_s          (raw HR rows!)
//   attnout = ctx @ Wv^T + bv
//   out     = LN( Q + attnout @ Wo^T + bo )
// => 4 GEMMs of [1024,768]@[768,768] (fp32 WMMA 16x16x4) + one streaming
//    pass over HR_image (201 MB -> ~8.6us at 23.3 TB/s, the roofline floor).
// All global->LDS staging uses CDNA5 async-to-LDS (ASYNCcnt) to bypass VGPRs.
// ---------------------------------------------------------------------------

#define EMB   768
#define NROW  1024          // B * 8*8*8 query/window count
#define WIN   64            // 4*4*4 tokens per window
#define XSTR  772           // padded LDS row stride (floats)
#define ASTR  772           // padded A-tile stride in GEMM LDS

typedef __attribute__((ext_vector_type(2))) float v2f;
typedef __attribute__((ext_vector_type(8))) float v8f;

// --- CDNA5 async global->LDS copy (GLOBAL_LOAD_ASYNC_TO_LDS_B128) ----------
// vdst = LDS byte address (low 32 bits of the generic pointer = wave-relative
// LDS offset), vaddr = 64-bit global address. Tracked with ASYNCcnt.
__device__ __forceinline__ void async_copy16(void* lds_ptr, const void* gptr) {
  const unsigned lds_off = (unsigned)(unsigned long long)lds_ptr;
  asm volatile("global_load_async_to_lds_b128 %0, %1, off"
               :: "v"(lds_off), "v"(gptr) : "memory");
}
__device__ __forceinline__ void async_wait_all() {
  asm volatile("s_wait_asynccnt 0x0" ::: "memory");
}

// ---------------------------------------------------------------------------
// Tiled fp32 GEMM with V_WMMA_F32_16X16X4_F32 (wave32).
//   C[m][n] = sum_k A[m][k] * B[k][n] (+ bias[n]) (+ resid[m][n])
//   TRANSB=1: B[k][n] = W[n*768 + k]   (A @ W^T, contiguous float2 loads)
//   TRANSB=0: B[k][n] = W[k*768 + n]
// Block = 128 threads = 4 waves. Block tile = 16(M) x 64(N); one 16x16 tile
// per wave. A 16x768 tile is async-staged in LDS and shared by all 4 waves.
//
// WMMA f32 16x16x4 fragment layout (ISA 7.12.2, wave32):
//   A (16x4):  lane -> M = lane%16 ; VGPR v holds K = v + 2*(lane>=16)
//   B (4x16):  lane -> N = lane%16 ; VGPR v holds K = v + 2*(lane>=16)
//   C/D:       lane -> N = lane%16 ; VGPR v holds M = v + 8*(lane>=16)
// ---------------------------------------------------------------------------
template <bool TRANSB, bool RESID>
__global__ __launch_bounds__(128)
void gemm768_wmma_f32(const float* __restrict__ A,
                      const float* __restrict__ W,
                      const float* __restrict__ bias,
                      const float* __restrict__ resid,
                      float* __restrict__ C) {
  __shared__ float Asm[16 * ASTR];           // 16 x 768 fp32 A tile (padded)

  const int t    = threadIdx.x;
  const int wave = t >> 5;
  const int lane = t & 31;
  const int m0   = blockIdx.x * 16;
  const int n0   = blockIdx.y * 64 + wave * 16;

  // Async-stage A tile (16 rows x 768) straight into LDS (no VGPR roundtrip).
  for (int i = t; i < 16 * (EMB / 4); i += 128) {
    const int rr = i / (EMB / 4);
    const int cc = i % (EMB / 4);
    async_copy16(Asm + rr * ASTR + cc * 4,
                 A + (size_t)(m0 + rr) * EMB + cc * 4);
  }
  async_wait_all();
  __syncthreads();

  const int mrow  = lane & 15;           // M index for A frag / N index for B frag
  const int khalf = (lane >> 4) * 2;     // K sub-offset per lane half

  v8f acc = {0.f, 0.f, 0.f, 0.f, 0.f, 0.f, 0.f, 0.f};

#pragma unroll 4
  for (int k = 0; k < EMB; k += 4) {
    // A fragment: contiguous pair from LDS (ds_load_b64)
    const float2 ap = *(const float2*)&Asm[mrow * ASTR + k + khalf];
    v2f a; a.x = ap.x; a.y = ap.y;

    v2f b;
    if (TRANSB) {
      const float2 bp =
          *(const float2*)&W[(size_t)(n0 + mrow) * EMB + k + khalf];
      b.x = bp.x; b.y = bp.y;
    } else {
      b.x = W[(size_t)(k + khalf)     * EMB + n0 + mrow];
      b.y = W[(size_t)(k + khalf + 1) * EMB + n0 + mrow];
    }

    // D = A(16x4) x B(4x16) + C    -> v_wmma_f32_16x16x4_f32
    acc = __builtin_amdgcn_wmma_f32_16x16x4_f32(
        /*neg_a=*/false, a, /*neg_b=*/false, b,
        /*c_mod=*/(short)0, acc, /*reuse_a=*/false, /*reuse_b=*/false);
  }

  // Epilogue: scatter 8 rows per lane, add bias / residual.
  const int mhalf = (lane >> 4) * 8;
  const int cn    = n0 + (lane & 15);
  const float bb  = bias ? bias[cn] : 0.f;
#pragma unroll
  for (int v = 0; v < 8; ++v) {
    const int cm = m0 + mhalf + v;
    float val = acc[v] + bb;
    if (RESID) val += resid[(size_t)cm * EMB + cn];
    C[(size_t)cm * EMB + cn] = val;
  }
}

// ---------------------------------------------------------------------------
// Per-window attention over raw HR rows. One workgroup per (b, d, w, h):
//   1) async-stage the 64x768 fp32 window + qk row into LDS (single HR read:
//      201 MB total -> memory-roofline-bound, ASYNCcnt path, no VGPR copies)
//   2) scores_s = scale * qk . x_s   (4 lanes per row, shfl reduce)
//   3) softmax over 64 (tiny, redundantly computed)
//   4) ctx[c] = sum_s a_s x_s[c]     (threads own 3 channels each)
// ---------------------------------------------------------------------------
__global__ __launch_bounds__(256)
void attn_window_kernel(const float* __restrict__ HR,
                        const float* __restrict__ QK,
                        float* __restrict__ ctx) {
  extern __shared__ float lds[];
  float* Xs = lds;                    // [64 * XSTR]
  float* qk = Xs + WIN * XSTR;        // [768]
  float* sc = qk + EMB;               // [64] raw scaled scores
  float* aw = sc + WIN;               // [64] exp weights

  const int n = blockIdx.x;           // 0..1023
  const int b = n >> 9;
  const int r = n & 511;
  const int d = r >> 6, w = (r >> 3) & 7, h = r & 7;
  const int t = threadIdx.x;

  // qk row -> LDS (async, 192 x b128)
  if (t < EMB / 4)
    async_copy16(qk + t * 4, QK + (size_t)n * EMB + t * 4);

  // window rows -> LDS (async, 4 threads per row, 48 x b128 each)
  const int s  = t >> 2;                  // 0..63
  const int l  = t & 3;
  const int sd = s >> 4, sw = (s >> 2) & 3, sh = s & 3;
  const float* grow =
      HR + ((((size_t)b * 32 + (d * 4 + sd)) * 32 + (w * 4 + sw)) * 32 +
            (h * 4 + sh)) * EMB;
  float* lrow = Xs + s * XSTR;
#pragma unroll 8
  for (int k = 0; k < EMB / 16; ++k) {    // 48 b128 per thread
    const int j = (l + k * 4) * 4;        // float offset
    async_copy16(lrow + j, grow + j);
  }
  async_wait_all();
  __syncthreads();

  // scores: each of the 4 lanes of a row covers 192 channels
  float part = 0.f;
#pragma unroll 8
  for (int k = 0; k < 48; ++k) {
    const float4 x = ((const float4*)lrow)[l * 48 + k];
    const float4 q = ((const float4*)qk)[l * 48 + k];
    part += x.x * q.x + x.y * q.y + x.z * q.z + x.w * q.w;
  }
  part += __shfl_xor(part, 1, 32);
  part += __shfl_xor(part, 2, 32);
  const float scale = 0.03608439182435161f;   // 1/sqrt(768)
  if (l == 0) sc[s] = part * scale;
  __syncthreads();

  // softmax over 64 (redundant per-thread; LDS broadcast reads)
  float mx = -3.4e38f;
#pragma unroll
  for (int i = 0; i < WIN; ++i) mx = fmaxf(mx, sc[i]);
  if (t < WIN) aw[t] = __expf(sc[t] - mx);
  __syncthreads();
  float sum = 0.f;
#pragma unroll
  for (int i = 0; i < WIN; ++i) sum += aw[i];
  const float inv = 1.f / sum;

  // weighted sum of raw rows: thread owns channels [3t, 3t+2]
  const int c0 = t * 3;
  float a0 = 0.f, a1 = 0.f, a2 = 0.f;
#pragma unroll 8
  for (int i = 0; i < WIN; ++i) {
    const float wgt = aw[i];
    const float* xr = Xs + i * XSTR + c0;
    a0 += wgt * xr[0];
    a1 += wgt * xr[1];
    a2 += wgt * xr[2];
  }
  float* op = ctx + (size_t)n * EMB + c0;
  op[0] = a0 * inv;
  op[1] = a1 * inv;
  op[2] = a2 * inv;
}

// ---------------------------------------------------------------------------
// LayerNorm over EMB=768 per row (block = 256 threads, 3 channels each).
// ---------------------------------------------------------------------------
__global__ __launch_bounds__(256)
void layernorm_kernel(const float* __restrict__ R,
                      const float* __restrict__ gamma,
                      const float* __restrict__ beta,
                      float* __restrict__ out) {
  __shared__ float s1[256];
  __shared__ float s2[256];
  const int row = blockIdx.x;
  const int t   = threadIdx.x;
  const int c   = t * 3;
  const float* rp = R + (size_t)row * EMB + c;
  const float v0 = rp[0], v1 = rp[1], v2 = rp[2];
  s1[t] = v0 + v1 + v2;
  s2[t] = v0 * v0 + v1 * v1 + v2 * v2;
  __syncthreads();
#pragma unroll
  for (int off = 128; off > 0; off >>= 1) {
    if (t < off) { s1[t] += s1[t + off]; s2[t] += s2[t + off]; }
    __syncthreads();
  }
  const float mean = s1[0] * (1.f / EMB);
  const float var  = s2[0] * (1.f / EMB) - mean * mean;
  const float inv  = rsqrtf(var + 1e-5f);
  float* op = out + (size_t)row * EMB + c;
  op[0] = (v0 - mean) * inv * gamma[c + 0] + beta[c + 0];
  op[1] = (v1 - mean) * inv * gamma[c + 1] + beta[c + 1];
  op[2] = (v2 - mean) * inv * gamma[c + 2] + beta[c + 2];
}

// ---------------------------------------------------------------------------
extern "C" void kernel_launch(void* const* d_in, const int* in_sizes, int n_in,
                              void* d_out, int out_size, void* d_ws, size_t ws_size,
                              hipStream_t stream) {
  const float* LR = (const float*)d_in[0];
  const float* HR = (const float*)d_in[1];
  const float* Wq = (const float*)d_in[2];
  const float* bq = (const float*)d_in[3];
  const float* Wk = (const float*)d_in[4];
  // d_in[5] = Wk_b : provably cancels in softmax, unused.
  const float* Wv = (const float*)d_in[6];
  const float* bv = (const float*)d_in[7];
  const float* Wo = (const float*)d_in[8];
  const float* bo = (const float*)d_in[9];
  const float* g  = (const float*)d_in[10];
  const float* bt = (const float*)d_in[11];

  float* buf0 = (float*)d_ws;                  // Qp   [1024,768]
  float* buf1 = buf0 + (size_t)NROW * EMB;     // QK -> Xv
  float* buf2 = buf1 + (size_t)NROW * EMB;     // ctx -> residual

  const dim3 gblk(128);
  const dim3 ggrd(NROW / 16, EMB / 64);

  // 1) Qp = LR @ Wq^T + bq
  gemm768_wmma_f32<true,  false><<<ggrd, gblk, 0, stream>>>(LR,   Wq, bq, nullptr, buf0);
  // 2) QK = Qp @ Wk
  gemm768_wmma_f32<false, false><<<ggrd, gblk, 0, stream>>>(buf0, Wk, nullptr, nullptr, buf1);
  // 3) ctx = softmax-attention over raw HR windows (single 201 MB stream)
  const size_t lds_bytes = (size_t)(WIN * XSTR + EMB + 2 * WIN) * sizeof(float);
  attn_window_kernel<<<dim3(NROW), dim3(256), lds_bytes, stream>>>(HR, buf1, buf2);
  // 4) Xv = ctx @ Wv^T + bv
  gemm768_wmma_f32<true,  false><<<ggrd, gblk, 0, stream>>>(buf2, Wv, bv, nullptr, buf1);
  // 5) R = Qp + Xv @ Wo^T + bo
  gemm768_wmma_f32<true,  true ><<<ggrd, gblk, 0, stream>>>(buf1, Wo, bo, buf0, buf2);
  // 6) out = LayerNorm(R) * g + b
  layernorm_kernel<<<dim3(NROW), dim3(256), 0, stream>>>(buf2, g, bt, (float*)d_out);
}